// GraphAttentionLayer_46514495816427
// MI455X (gfx1250) — compile-verified
//
#include <hip/hip_runtime.h>
#include <math.h>

typedef __attribute__((ext_vector_type(2))) float v2f;
typedef __attribute__((ext_vector_type(8))) float v8f;

#define GAT_ALPHA 0.2f
#define GAT_NEG_INF -9.0e15f

// LDS offset of a generic pointer to __shared__ (flat aperture: low 32 bits).
__device__ __forceinline__ unsigned lds_off(const void* p) {
  return (unsigned)(size_t)p;
}

__device__ __forceinline__ void async_load_b128(unsigned lds_dst, const void* gsrc) {
  // GLOBAL_LOAD_ASYNC_TO_LDS_B128, GV mode (64-bit vaddr, saddr=off).
  // Tracked with ASYNCcnt; LDS write needs no VGPR round-trip.
  asm volatile("global_load_async_to_lds_b128 %0, %1, off"
               :: "v"(lds_dst), "v"(gsrc) : "memory");
}

__device__ __forceinline__ void wait_async0() {
  asm volatile("s_wait_asynccnt 0x0" ::: "memory");
}

// ---------------------------------------------------------------------------
// Tiled fp32 GEMM using V_WMMA_F32_16X16X4_F32 with double-buffered LDS tiles
// filled by GLOBAL_LOAD_ASYNC_TO_LDS_B128.
//   C[M,N] = A[M,K] * B      (BT=false: B is [K,N] row-major)
//   C[M,N] = A[M,K] * B^T    (BT=true : B is [N,K] row-major, torch Linear W)
// Block tile: 64(M) x 32(N), K-step 16. 256 threads = 8 waves, each wave owns
// one 16x16 output tile.
// ---------------------------------------------------------------------------
template <bool BT>
__global__ __launch_bounds__(256) void gemm_wmma(const float* __restrict__ A,
                                                 const float* __restrict__ B,
                                                 float* __restrict__ C,
                                                 int M, int N, int K) {
  // Pad to 20/36 floats per row: 16B-aligned rows for async B128 into LDS,
  // and bank-conflict-free column reads (20*m mod 64 and 36*k mod 64 spread).
  __shared__ float As[2][64][20];
  __shared__ float Bs[2][16][36];

  const int tid  = threadIdx.x;
  const int wave = tid >> 5;   // 0..7
  const int lane = tid & 31;
  const int bm   = blockIdx.y * 64;  // M block (slow grid dim)
  const int bn   = blockIdx.x * 32;  // N block (fast grid dim -> L2 reuse of A)
  const int wm   = (wave & 3) * 16;
  const int wn   = (wave >> 2) * 16;
  const int m    = lane & 15;            // row within 16x16 fragment
  const int kh   = (lane >> 4) << 1;     // K-pair select: 0 or 2

  // Stage one 64x16 A tile + 16x32 B tile into buffer `buf` (async where possible).
  auto stage = [&](int buf, int k0) {
    {  // A tile: 256 lanes x float4, coalesced async B128 -> LDS
      const int r = tid >> 2;
      const int c = (tid & 3) << 2;
      async_load_b128(lds_off(&As[buf][r][c]),
                      A + (size_t)(bm + r) * K + k0 + c);
    }
    if (BT) {
      // B = W [N x K]: transpose through VGPRs (async cannot transpose).
      if (tid < 128) {
        const int nn = tid >> 2;
        const int kk = (tid & 3) << 2;
        const float4 v = *(const float4*)(B + (size_t)(bn + nn) * K + k0 + kk);
        Bs[buf][kk + 0][nn] = v.x; Bs[buf][kk + 1][nn] = v.y;
        Bs[buf][kk + 2][nn] = v.z; Bs[buf][kk + 3][nn] = v.w;
      }
    } else {
      // B [K x N]: rows contiguous -> async B128, lanes 128..255.
      if (tid >= 128) {
        const int u  = tid - 128;
        const int kk = u >> 3;
        const int nn = (u & 7) << 2;
        async_load_b128(lds_off(&Bs[buf][kk][nn]),
                        B + (size_t)(k0 + kk) * N + bn + nn);
      }
    }
  };

  v8f acc = {};

  stage(0, 0);
  wait_async0();
  __syncthreads();

  int cur = 0;
  for (int k0 = 0; k0 < K; k0 += 16) {
    if (k0 + 16 < K) stage(cur ^ 1, k0 + 16);  // prefetch next tile (overlapped)

#pragma unroll
    for (int s = 0; s < 4; ++s) {
      const int kb = (s << 2) + kh;
      v2f a, b;
      // A 16x4 fragment: lane m holds K=kb,kb+1 in VGPR0/1 (lanes 16-31: +2)
      a.x = As[cur][wm + m][kb];
      a.y = As[cur][wm + m][kb + 1];
      // B 4x16 fragment: mirrored layout
      b.x = Bs[cur][kb][wn + m];
      b.y = Bs[cur][kb + 1][wn + m];
      acc = __builtin_amdgcn_wmma_f32_16x16x4_f32(
          /*neg_a=*/false, a, /*neg_b=*/false, b,
          /*c_mod=*/(short)0, acc, /*reuse_a=*/false, /*reuse_b=*/false);
    }

    wait_async0();     // next buffer fully in LDS (this wave's asyncs done)
    __syncthreads();   // all waves done filling + done reading `cur`
    cur ^= 1;
  }

  // C/D layout: VGPR r -> row r (lanes 0-15) / row r+8 (lanes 16-31)
  const int row0 = bm + wm + ((lane >> 4) << 3);
  const int col  = bn + wn + m;
#pragma unroll
  for (int r = 0; r < 8; ++r)
    C[(size_t)(row0 + r) * N + col] = acc[r];
}

// ---------------------------------------------------------------------------
// s_src[i] = Wh[i,:].a[:F];  s_dst[i] = Wh[i,:].a[F:].  One wave32 per row.
// ---------------------------------------------------------------------------
__global__ __launch_bounds__(256) void dots_kernel(const float* __restrict__ Wh,
                                                   const float* __restrict__ a_vec,
                                                   float* __restrict__ s_src,
                                                   float* __restrict__ s_dst,
                                                   int N, int F) {
  const int row  = blockIdx.x * 8 + (threadIdx.x >> 5);
  const int lane = threadIdx.x & 31;
  const float* wr = Wh + (size_t)row * F;
  float s1 = 0.f, s2 = 0.f;
  for (int f = lane; f < F; f += 32) {
    const float w = wr[f];
    s1 += w * a_vec[f];
    s2 += w * a_vec[F + f];
  }
#pragma unroll
  for (int off = 16; off > 0; off >>= 1) {
    s1 += __shfl_xor(s1, off, 32);
    s2 += __shfl_xor(s2, off, 32);
  }
  if (lane == 0) { s_src[row] = s1; s_dst[row] = s2; }
}

// ---------------------------------------------------------------------------
// One 256-thread block per row i: masked leaky-relu scores -> softmax -> att.
// 32 columns per thread kept in registers (single pass over adj row).
// ---------------------------------------------------------------------------
__global__ __launch_bounds__(256) void attention_kernel(const float* __restrict__ adj,
                                                        const float* __restrict__ s_src,
                                                        const float* __restrict__ s_dst,
                                                        float* __restrict__ att,
                                                        int N) {
  const int i   = blockIdx.x;
  const int tid = threadIdx.x;
  const float si = s_src[i];
  const float* arow = adj + (size_t)i * N;

  float sc[32];
  float lmax = GAT_NEG_INF;
#pragma unroll
  for (int t = 0; t < 32; ++t) {
    const int j = tid + (t << 8);            // coalesced
    if ((t & 7) == 0)                        // stream-ahead hint (8KB)
      __builtin_prefetch(arow + j + 2048, 0, 1);
    const float aj = arow[j];
    const float x  = si + s_dst[j];
    const float e  = x > 0.f ? x : GAT_ALPHA * x;
    const float s  = aj > 0.f ? e : GAT_NEG_INF;
    sc[t] = s;
    lmax = fmaxf(lmax, s);
  }

  __shared__ float red[8];
#pragma unroll
  for (int off = 16; off > 0; off >>= 1) lmax = fmaxf(lmax, __shfl_xor(lmax, off, 32));
  if ((tid & 31) == 0) red[tid >> 5] = lmax;
  __syncthreads();
  const float bmax = fmaxf(fmaxf(fmaxf(red[0], red[1]), fmaxf(red[2], red[3])),
                           fmaxf(fmaxf(red[4], red[5]), fmaxf(red[6], red[7])));

  float lsum = 0.f;
#pragma unroll
  for (int t = 0; t < 32; ++t) {
    const float p = __expf(sc[t] - bmax);    // fully-masked row -> uniform, like ref
    sc[t] = p;
    lsum += p;
  }
#pragma unroll
  for (int off = 16; off > 0; off >>= 1) lsum += __shfl_xor(lsum, off, 32);
  __syncthreads();  // red reads above complete before rewrite
  if ((tid & 31) == 0) red[tid >> 5] = lsum;
  __syncthreads();
  const float inv = 1.0f / (red[0] + red[1] + red[2] + red[3] +
                            red[4] + red[5] + red[6] + red[7]);

  float* orow = att + (size_t)i * N;
#pragma unroll
  for (int t = 0; t < 32; ++t) {
    const int j = tid + (t << 8);
    orow[j] = sc[t] * inv;
  }
}

// ---------------------------------------------------------------------------
extern "C" void kernel_launch(void* const* d_in, const int* in_sizes, int n_in,
                              void* d_out, int out_size, void* d_ws, size_t ws_size,
                              hipStream_t stream) {
  const float* h     = (const float*)d_in[0];  // [8192, 512]
  const float* adj   = (const float*)d_in[1];  // [8192, 8192]
  const float* W     = (const float*)d_in[2];  // [256, 512]
  const float* a_vec = (const float*)d_in[3];  // [512]

  const int N = 8192, IN_F = 512, OUT_F = 256;

  float* h_prime = (float*)d_out;                     // [8192, 256]
  float* att     = (float*)d_out + (size_t)N * OUT_F; // [8192, 8192]

  float* Wh    = (float*)d_ws;                        // [8192, 256] = 8 MB
  float* s_src = Wh + (size_t)N * OUT_F;              // [8192]
  float* s_dst = s_src + N;                           // [8192]

  // 1) Wh = h @ W^T   (M=8192, N=256, K=512)
  {
    dim3 grid(OUT_F / 32, N / 64);
    gemm_wmma<true><<<grid, 256, 0, stream>>>(h, W, Wh, N, OUT_F, IN_F);
  }
  // 2) per-row attention logits
  dots_kernel<<<N / 8, 256, 0, stream>>>(Wh, a_vec, s_src, s_dst, N, OUT_F);
  // 3) masked softmax rows -> attention output
  attention_kernel<<<N, 256, 0, stream>>>(adj, s_src, s_dst, att, N);
  // 4) h_prime = attention @ Wh   (M=8192, N=256, K=8192)
  {
    dim3 grid(OUT_F / 32, N / 64);
    gemm_wmma<false><<<grid, 256, 0, stream>>>(att, Wh, h_prime, N, OUT_F, N);
  }
}